// GeneGraphAttentionEncoder_21792664060256
// MI455X (gfx1250) — compile-verified
//
#include <hip/hip_runtime.h>
#include <hip/hip_bf16.h>

#define NROWS 16384
#define G 100
#define C 10
#define H 32
#define S 10
#define GPC 10          // genes per chromosome
#define BN_EPS 1e-5f
#define SLOPE 0.2f

#define ROWS_PER_WAVE 16
#define WAVES_PER_BLOCK 8
#define ROWS_PER_BLOCK (ROWS_PER_WAVE * WAVES_PER_BLOCK)   // 128
#define BLOCK_THREADS 256
#define KCHUNKS (H / 4)                                    // 8 WMMA steps

typedef __attribute__((ext_vector_type(2))) float v2f;
typedef __attribute__((ext_vector_type(8))) float v8f;

__device__ __forceinline__ float waveSum32(float v) {
    #pragma unroll
    for (int off = 16; off > 0; off >>= 1)
        v += __shfl_xor(v, off, 32);
    return v;
}

__global__ void gga_zero_kernel(float* sums) {
    int i = threadIdx.x;
    if (i < 2 * S) sums[i] = 0.0f;
}

__global__ __launch_bounds__(BLOCK_THREADS)
void gga_main_kernel(const int* __restrict__ x,
                     const float* __restrict__ emb,        // (G,3,H)
                     const float* __restrict__ gene_att,   // (C,H)
                     const float* __restrict__ chrom_att,  // (1,H)
                     const float* __restrict__ W,          // (H,S)
                     const float* __restrict__ bvec,       // (S)
                     float* __restrict__ y,                // (N,S) in ws
                     float* __restrict__ sums)             // [S]=sum, [S..2S)=sumsq
{
    __shared__ float s_emb[G * 3 * H];                        // 38400 B
    __shared__ float s_gatt[C * H];                           //  1280 B
    __shared__ float s_catt[H];                               //   128 B
    // B operand of V_WMMA_F32_16X16X4_F32, pre-padded & pre-swizzled:
    // s_B[kc][lane] = { W[4kc+2*half + 0][col], W[4kc+2*half + 1][col] }, col>=S -> 0
    __shared__ v2f   s_B[KCHUNKS * 32];                       //  2048 B
    __shared__ float s_b[S];
    __shared__ float s_g[WAVES_PER_BLOCK][ROWS_PER_WAVE][H];  // 16384 B

    const int tid = threadIdx.x;
    for (int i = tid; i < G * 3 * H; i += BLOCK_THREADS) s_emb[i] = emb[i];
    for (int i = tid; i < C * H;     i += BLOCK_THREADS) s_gatt[i] = gene_att[i];
    for (int i = tid; i < H;         i += BLOCK_THREADS) s_catt[i] = chrom_att[i];
    for (int i = tid; i < KCHUNKS * 32 * 2; i += BLOCK_THREADS) {
        const int kc     = i >> 6;
        const int lane_i = (i >> 1) & 31;
        const int j      = i & 1;
        const int half_i = lane_i >> 4;
        const int col    = lane_i & 15;
        const int k      = 4 * kc + 2 * half_i + j;
        ((float*)s_B)[i] = (col < S) ? W[k * S + col] : 0.0f;
    }
    for (int i = tid; i < S;         i += BLOCK_THREADS) s_b[i] = bvec[i];
    __syncthreads();

    const int wave = tid >> 5;
    const int lane = tid & 31;   // lane == hidden index h
    const int rowbase = blockIdx.x * ROWS_PER_BLOCK + wave * ROWS_PER_WAVE;

    // ---- per-row gene/chrom attention: whole wave works one row at a time ----
    for (int r = 0; r < ROWS_PER_WAVE; ++r) {
        const int n = rowbase + r;
        float chrom[C];
        #pragma unroll
        for (int c = 0; c < C; ++c) {
            float acc = 0.0f, asum = 0.0f;
            const float ga = s_gatt[c * H + lane];
            #pragma unroll
            for (int k = 0; k < GPC; ++k) {
                const int g = c * GPC + k;
                const int xv = x[n * G + g];                 // uniform across wave
                const float hv = s_emb[(g * 3 + xv) * H + lane];
                float att = waveSum32(ga * hv);              // gene_att[c]·h[n,g]
                const float lr = att > 0.0f ? att : SLOPE * att;
                const float a  = (att != 0.0f) ? __expf(lr) : 0.0f;
                acc  += a * hv;
                asum += a;
            }
            const float ch = acc / fmaxf(asum, 1e-10f);
            chrom[c] = fmaxf(ch, 0.0f);                      // relu
        }
        // chromosome-level attention
        float ca[C];
        float casum = 0.0f;
        const float cav = s_catt[lane];
        #pragma unroll
        for (int c = 0; c < C; ++c) {
            float catt = waveSum32(cav * chrom[c]);
            const float lr = catt > 0.0f ? catt : SLOPE * catt;
            ca[c] = __expf(lr);
            casum += ca[c];
        }
        float gv = 0.0f;
        const float inv = 1.0f / casum;
        #pragma unroll
        for (int c = 0; c < C; ++c) gv += (ca[c] * inv) * chrom[c];
        gv = fmaxf(gv, 0.0f);                                // relu
        s_g[wave][r][lane] = gv;
    }
    __syncthreads();

    // ---- y[16x10] = g_tile(16x32) @ W(32x10) + b  via V_WMMA_F32_16X16X4_F32 ----
    // A 16x4 f32 layout: lanes 0-15 rows M=0..15 (VGPR0:K=0, VGPR1:K=1);
    //                    lanes 16-31 same rows   (VGPR0:K=2, VGPR1:K=3).
    // B pre-swizzled in s_B: one unconditional ds_load_b64 per operand.
    const int half = lane >> 4;      // 0 or 1
    const int lid  = lane & 15;      // row for A, col for B/C
    v8f acc = {0.f, 0.f, 0.f, 0.f, 0.f, 0.f, 0.f, 0.f};
    #pragma unroll
    for (int kc = 0; kc < KCHUNKS; ++kc) {
        const int k0 = 4 * kc + 2 * half;
        const v2f a = *(const v2f*)&s_g[wave][lid][k0];      // 8B-aligned (k0 even)
        const v2f b = s_B[kc * 32 + lane];
        acc = __builtin_amdgcn_wmma_f32_16x16x4_f32(
            /*neg_a=*/false, a, /*neg_b=*/false, b,
            /*c_mod=*/(short)0, acc, /*reuse_a=*/false, /*reuse_b=*/false);
    }

    // C/D layout: VGPR r -> (M=r, N=lid) for lanes 0-15, (M=r+8, N=lid) for 16-31.
    float lsum = 0.0f, lsq = 0.0f;
    #pragma unroll
    for (int r8 = 0; r8 < 8; ++r8) {
        const int row = rowbase + r8 + 8 * half;
        if (lid < S) {
            const float val = acc[r8] + s_b[lid];
            y[row * S + lid] = val;
            lsum += val;
            lsq  += val * val;
        }
    }
    if (lid < S) {
        atomicAdd(&sums[lid],     lsum);
        atomicAdd(&sums[S + lid], lsq);
    }
}

__global__ void gga_bn_kernel(const float* __restrict__ y,
                              const float* __restrict__ sums,
                              const float* __restrict__ gamma,
                              const float* __restrict__ beta,
                              float* __restrict__ out)
{
    const int idx = blockIdx.x * blockDim.x + threadIdx.x;
    if (idx >= NROWS * S) return;
    const int col = idx % S;
    const float invN = 1.0f / (float)NROWS;
    const float mu   = sums[col] * invN;
    const float var  = sums[S + col] * invN - mu * mu;
    const float rstd = rsqrtf(var + BN_EPS);
    out[idx] = (y[idx] - mu) * rstd * gamma[col] + beta[col];
}

extern "C" void kernel_launch(void* const* d_in, const int* in_sizes, int n_in,
                              void* d_out, int out_size, void* d_ws, size_t ws_size,
                              hipStream_t stream) {
    const int*   x         = (const int*)  d_in[0];
    const float* emb       = (const float*)d_in[1];
    const float* gene_att  = (const float*)d_in[2];
    const float* chrom_att = (const float*)d_in[3];
    const float* W         = (const float*)d_in[4];
    const float* b         = (const float*)d_in[5];
    const float* gamma     = (const float*)d_in[6];
    const float* beta      = (const float*)d_in[7];

    float* y    = (float*)d_ws;          // N*S floats
    float* sums = y + NROWS * S;         // 2*S floats (sum, sumsq)

    gga_zero_kernel<<<1, 32, 0, stream>>>(sums);
    gga_main_kernel<<<NROWS / ROWS_PER_BLOCK, BLOCK_THREADS, 0, stream>>>(
        x, emb, gene_att, chrom_att, W, b, y, sums);
    gga_bn_kernel<<<(NROWS * S + 255) / 256, 256, 0, stream>>>(
        y, sums, gamma, beta, (float*)d_out);
}